// RmiModel_22737556865378
// MI455X (gfx1250) — compile-verified
//
#include <hip/hip_runtime.h>
#include <hip/hip_bf16.h>

// Problem constants (match reference)
#define BATCH 16384
#define NTIME 200
#define NSEG  8
#define CHUNK 25          // ceil(199/8); last segment has 24 steps
#define TPB1  64          // threads per block, kernel 1 (2 wave32)
#define LSTRIDE 185       // per-lane LDS floats: 7*26=182 (+3 pad); gcd(185,64)=1

// --- gfx1250 async global->LDS path (guarded; falls back to sync copy) -----
#if defined(__has_builtin)
#  if __has_builtin(__builtin_amdgcn_global_load_async_to_lds_b32)
#    define HAVE_ASYNC_LDS 1
#  endif
#  if __has_builtin(__builtin_amdgcn_s_wait_asynccnt)
#    define HAVE_WAIT_ASYNC 1
#  endif
#endif

typedef __attribute__((address_space(1))) int* as1_i32p;
typedef __attribute__((address_space(3))) int* as3_i32p;

// Stage one row of W contiguous b32 elements global->LDS. The instruction
// IOFFSET applies to BOTH addresses, so a
// whole row shares one base VGPR pair and differs only in the immediate.
// Template recursion guarantees the offset is an ICE for the builtin.
template <int KK, int W>
struct StageRow {
    static __device__ __forceinline__ void run(const float* g, float* l) {
#ifdef HAVE_ASYNC_LDS
        __builtin_amdgcn_global_load_async_to_lds_b32(
            (as1_i32p)g, (as3_i32p)l, KK * 4 /*imm offset*/, 0 /*cpol*/);
#else
        l[KK] = g[KK];   // fallback: global load + ds_store
#endif
        StageRow<KK + 1, W>::run(g, l);
    }
};
template <int W>
struct StageRow<W, W> {
    static __device__ __forceinline__ void run(const float*, float*) {}
};

template <int W>
__device__ __forceinline__ void stage_all_rows(const float* xb, float* lb) {
    #pragma unroll
    for (int c = 0; c < 7; ++c)
        StageRow<0, W>::run(xb + c * NTIME, lb + c * 26);
}

__device__ __forceinline__ void wait_async_done() {
#ifdef HAVE_WAIT_ASYNC
    __builtin_amdgcn_s_wait_asynccnt(0);
#endif
    __asm__ volatile("" ::: "memory");   // keep LDS reads after the wait
}

// ---------------------------------------------------------------------------
// Kernel 1: integrate one segment of the preintegration scan per thread.
// Block = 64 lanes = 64 consecutive batches, all with the same segment s.
// Phase 1: each lane async-stages its own 7xW input window into LDS.
// Phase 2: s_wait_asynccnt 0, then run the 25-step scan out of LDS.
// ---------------------------------------------------------------------------
__global__ __launch_bounds__(TPB1) void rmi_segment_kernel(
    const float* __restrict__ x,   // (B, 7, N) contiguous
    float* __restrict__ seg)       // (NSEG, B, 16) floats
{
    __shared__ float smem[TPB1 * LSTRIDE];   // 47,360 B

    const int lane = threadIdx.x;
    const int gid  = blockIdx.x * TPB1 + lane;   // tid = s*BATCH + b
    const int b    = gid & (BATCH - 1);
    const int s    = gid >> 14;                  // BATCH == 2^14

    const int k0 = s * CHUNK;

    const float* __restrict__ xb = x + (size_t)b * (7 * NTIME) + k0;
    float* __restrict__ lb = smem + lane * LSTRIDE;

    // gfx1250 global_prefetch_b8 on each channel stream
    __builtin_prefetch(xb + 0 * NTIME, 0, 3);
    __builtin_prefetch(xb + 1 * NTIME, 0, 3);
    __builtin_prefetch(xb + 2 * NTIME, 0, 3);
    __builtin_prefetch(xb + 3 * NTIME, 0, 3);
    __builtin_prefetch(xb + 4 * NTIME, 0, 3);
    __builtin_prefetch(xb + 5 * NTIME, 0, 3);
    __builtin_prefetch(xb + 6 * NTIME, 0, 3);

    // Stage this lane's 7 channel windows into LDS (async, no VGPR roundtrip)
    int nsteps;
    if (s < NSEG - 1) { stage_all_rows<26>(xb, lb); nsteps = 25; }
    else              { stage_all_rows<25>(xb, lb); nsteps = 24; }
    wait_async_done();

    const float* __restrict__ tc = lb + 0 * 26;
    const float* __restrict__ gx = lb + 1 * 26;
    const float* __restrict__ gy = lb + 2 * 26;
    const float* __restrict__ gz = lb + 3 * 26;
    const float* __restrict__ ax = lb + 4 * 26;
    const float* __restrict__ ay = lb + 5 * 26;
    const float* __restrict__ az = lb + 6 * 26;

    // Segment-local state: C = I, V = 0, R = 0
    float C00 = 1.f, C01 = 0.f, C02 = 0.f;
    float C10 = 0.f, C11 = 1.f, C12 = 0.f;
    float C20 = 0.f, C21 = 0.f, C22 = 1.f;
    float V0 = 0.f, V1 = 0.f, V2 = 0.f;
    float R0 = 0.f, R1 = 0.f, R2 = 0.f;

    const float t_start = tc[0];
    float tprev = t_start;

    for (int kk = 0; kk < nsteps; ++kk) {
        const float tn = tc[kk + 1];
        const float d  = tn - tprev;
        tprev = tn;

        const float a0 = ax[kk], a1 = ay[kk], a2 = az[kk];
        const float wx = gx[kk] * d, wy = gy[kk] * d, wz = gz[kk] * d;

        // Ca = C * a
        const float Ca0 = fmaf(C00, a0, fmaf(C01, a1, C02 * a2));
        const float Ca1 = fmaf(C10, a0, fmaf(C11, a1, C12 * a2));
        const float Ca2 = fmaf(C20, a0, fmaf(C21, a1, C22 * a2));

        const float hd2 = 0.5f * d * d;
        R0 = fmaf(V0, d, fmaf(Ca0, hd2, R0));
        R1 = fmaf(V1, d, fmaf(Ca1, hd2, R1));
        R2 = fmaf(V2, d, fmaf(Ca2, hd2, R2));
        V0 = fmaf(Ca0, d, V0);
        V1 = fmaf(Ca1, d, V1);
        V2 = fmaf(Ca2, d, V2);

        // so3_exp(phi), phi = w*d.  W^2 = phi*phi^T - th2*I
        const float th2  = fmaf(wx, wx, fmaf(wy, wy, wz * wz));
        const bool small = th2 < 1e-12f;
        const float th2s = small ? 1.0f : th2;
        const float th   = __fsqrt_rn(th2s);
        const float A  = small ? (1.0f - th2 * (1.0f / 6.0f))
                               : __fdividef(__sinf(th), th);
        const float Bc = small ? (0.5f - th2 * (1.0f / 24.0f))
                               : __fdividef(1.0f - __cosf(th), th2s);

        const float Bxy = Bc * wx * wy;
        const float Bxz = Bc * wx * wz;
        const float Byz = Bc * wy * wz;
        const float E00 = fmaf(Bc, fmaf(wx, wx, -th2), 1.0f);
        const float E11 = fmaf(Bc, fmaf(wy, wy, -th2), 1.0f);
        const float E22 = fmaf(Bc, fmaf(wz, wz, -th2), 1.0f);
        const float E01 = fmaf(-A, wz, Bxy);
        const float E10 = fmaf( A, wz, Bxy);
        const float E02 = fmaf( A, wy, Bxz);
        const float E20 = fmaf(-A, wy, Bxz);
        const float E12 = fmaf(-A, wx, Byz);
        const float E21 = fmaf( A, wx, Byz);

        // C = C * E
        const float n00 = fmaf(C00, E00, fmaf(C01, E10, C02 * E20));
        const float n01 = fmaf(C00, E01, fmaf(C01, E11, C02 * E21));
        const float n02 = fmaf(C00, E02, fmaf(C01, E12, C02 * E22));
        const float n10 = fmaf(C10, E00, fmaf(C11, E10, C12 * E20));
        const float n11 = fmaf(C10, E01, fmaf(C11, E11, C12 * E21));
        const float n12 = fmaf(C10, E02, fmaf(C11, E12, C12 * E22));
        const float n20 = fmaf(C20, E00, fmaf(C21, E10, C22 * E20));
        const float n21 = fmaf(C20, E01, fmaf(C21, E11, C22 * E21));
        const float n22 = fmaf(C20, E02, fmaf(C21, E12, C22 * E22));
        C00 = n00; C01 = n01; C02 = n02;
        C10 = n10; C11 = n11; C12 = n12;
        C20 = n20; C21 = n21; C22 = n22;
    }

    const float T = tprev - t_start;   // segment duration

    // Coalesced b128 stores: lane b writes 64 contiguous bytes.
    float4* __restrict__ w4 = (float4*)(seg + ((size_t)s * BATCH + b) * 16);
    w4[0] = make_float4(R0, R1, R2, V0);
    w4[1] = make_float4(V1, V2, C00, C01);
    w4[2] = make_float4(C02, C10, C11, C12);
    w4[3] = make_float4(C20, C21, C22, T);
}

// ---------------------------------------------------------------------------
// Kernel 2: compose the 8 segment states left-to-right per batch and emit
// [DR(3), DV(3), DC(9 row-major)] = 15 floats per batch.
// Composition: P' = P∘Q :  R += V*T_Q + C*R_Q ;  V += C*V_Q ;  C = C*C_Q
// ---------------------------------------------------------------------------
__global__ __launch_bounds__(256) void rmi_combine_kernel(
    const float* __restrict__ seg,
    float* __restrict__ out)
{
    const int b = blockIdx.x * blockDim.x + threadIdx.x;

    float C00 = 1.f, C01 = 0.f, C02 = 0.f;
    float C10 = 0.f, C11 = 1.f, C12 = 0.f;
    float C20 = 0.f, C21 = 0.f, C22 = 1.f;
    float V0 = 0.f, V1 = 0.f, V2 = 0.f;
    float R0 = 0.f, R1 = 0.f, R2 = 0.f;

    for (int s = 0; s < NSEG; ++s) {
        const float4* __restrict__ w4 =
            (const float4*)(seg + ((size_t)s * BATCH + b) * 16);
        const float4 q0 = w4[0], q1 = w4[1], q2 = w4[2], q3 = w4[3];
        const float Rs0 = q0.x, Rs1 = q0.y, Rs2 = q0.z;
        const float Vs0 = q0.w, Vs1 = q1.x, Vs2 = q1.y;
        const float S00 = q1.z, S01 = q1.w, S02 = q2.x;
        const float S10 = q2.y, S11 = q2.z, S12 = q2.w;
        const float S20 = q3.x, S21 = q3.y, S22 = q3.z;
        const float Ts  = q3.w;

        R0 = fmaf(V0, Ts, R0) + fmaf(C00, Rs0, fmaf(C01, Rs1, C02 * Rs2));
        R1 = fmaf(V1, Ts, R1) + fmaf(C10, Rs0, fmaf(C11, Rs1, C12 * Rs2));
        R2 = fmaf(V2, Ts, R2) + fmaf(C20, Rs0, fmaf(C21, Rs1, C22 * Rs2));

        V0 += fmaf(C00, Vs0, fmaf(C01, Vs1, C02 * Vs2));
        V1 += fmaf(C10, Vs0, fmaf(C11, Vs1, C12 * Vs2));
        V2 += fmaf(C20, Vs0, fmaf(C21, Vs1, C22 * Vs2));

        const float n00 = fmaf(C00, S00, fmaf(C01, S10, C02 * S20));
        const float n01 = fmaf(C00, S01, fmaf(C01, S11, C02 * S21));
        const float n02 = fmaf(C00, S02, fmaf(C01, S12, C02 * S22));
        const float n10 = fmaf(C10, S00, fmaf(C11, S10, C12 * S20));
        const float n11 = fmaf(C10, S01, fmaf(C11, S11, C12 * S21));
        const float n12 = fmaf(C10, S02, fmaf(C11, S12, C12 * S22));
        const float n20 = fmaf(C20, S00, fmaf(C21, S10, C22 * S20));
        const float n21 = fmaf(C20, S01, fmaf(C21, S11, C22 * S21));
        const float n22 = fmaf(C20, S02, fmaf(C21, S12, C22 * S22));
        C00 = n00; C01 = n01; C02 = n02;
        C10 = n10; C11 = n11; C12 = n12;
        C20 = n20; C21 = n21; C22 = n22;
    }

    float* __restrict__ o = out + (size_t)b * 15;
    o[0]  = R0;  o[1]  = R1;  o[2]  = R2;
    o[3]  = V0;  o[4]  = V1;  o[5]  = V2;
    o[6]  = C00; o[7]  = C01; o[8]  = C02;
    o[9]  = C10; o[10] = C11; o[11] = C12;
    o[12] = C20; o[13] = C21; o[14] = C22;
}

extern "C" void kernel_launch(void* const* d_in, const int* in_sizes, int n_in,
                              void* d_out, int out_size, void* d_ws, size_t ws_size,
                              hipStream_t stream) {
    const float* x = (const float*)d_in[0];   // (16384, 7, 200) float32
    float* out = (float*)d_out;               // (16384, 15) float32
    float* seg = (float*)d_ws;                // (8, 16384, 16) float32 = 8 MB

    (void)in_sizes; (void)n_in; (void)out_size; (void)ws_size;

    rmi_segment_kernel<<<(BATCH * NSEG) / TPB1, TPB1, 0, stream>>>(x, seg);
    rmi_combine_kernel<<<BATCH / 256, 256, 0, stream>>>(seg, out);
}